// DPCWrapper_19688130085713
// MI455X (gfx1250) — compile-verified
//
#include <hip/hip_runtime.h>
#include <cstdint>

// Problem constants (match reference)
#define BN 64
#define AN 8400
#define CN 80
#define KN 300
#define HFN 160
#define NSORT 512

typedef float vfloat4 __attribute__((ext_vector_type(4)));
typedef float v2f __attribute__((ext_vector_type(2)));
typedef float v8f __attribute__((ext_vector_type(8)));

#if __has_builtin(__builtin_amdgcn_wmma_f32_16x16x4_f32)
#define USE_WMMA_SCAN 1
#else
#define USE_WMMA_SCAN 0
#endif

// ---------------------------------------------------------------------------
// Kernel 1: per-anchor scoring. Streams cls_logits (172 MB, single use) with
// non-temporal b128 loads; field (100 KB/batch, heavy reuse) stays in L2.
// Emits order-preserving uint32 key of the masked combined score, best class,
// and beta_small into workspace.
// ---------------------------------------------------------------------------
__global__ __launch_bounds__(256) void dpc_score(
    const float* __restrict__ boxes, const float* __restrict__ obj,
    const float* __restrict__ cls, const float* __restrict__ field,
    unsigned* __restrict__ okey, int* __restrict__ oclsidx,
    float* __restrict__ obeta)
{
  const int a = blockIdx.x * blockDim.x + threadIdx.x;
  const int b = blockIdx.y;
  if (a >= AN) return;
  const long base = (long)b * AN + a;

  vfloat4 bx = *(const vfloat4*)(boxes + 4 * base);
  const float x1 = bx.x, y1 = bx.y, x2 = bx.z, y2 = bx.w;
  const float* __restrict__ F = field + (long)b * (HFN * HFN);

  // 3x3 grid of bilinear samples, mean -> beta
  float beta = 0.0f;
#pragma unroll
  for (int i = 0; i < 3; ++i) {
    const float ti = (i + 0.5f) / 3.0f;           // constant-folded, exact div
    const float ys = y1 + ti * (y2 - y1);
    const float v  = fminf(fmaxf(ys * 0.25f - 0.5f, 0.0f), 159.0f);
    const float v0f = floorf(v);
    const float fv  = v - v0f;
    const int v0 = (int)v0f;
    const int v1 = min(v0 + 1, HFN - 1);
#pragma unroll
    for (int j = 0; j < 3; ++j) {
      const float tj = (j + 0.5f) / 3.0f;
      const float xs = x1 + tj * (x2 - x1);
      const float u  = fminf(fmaxf(xs * 0.25f - 0.5f, 0.0f), 159.0f);
      const float u0f = floorf(u);
      const float fu  = u - u0f;
      const int u0 = (int)u0f;
      const int u1 = min(u0 + 1, HFN - 1);
      const float f00 = F[v0 * HFN + u0];
      const float f01 = F[v0 * HFN + u1];
      const float f10 = F[v1 * HFN + u0];
      const float f11 = F[v1 * HFN + u1];
      beta += f00 * (1.0f - fv) * (1.0f - fu) + f01 * (1.0f - fv) * fu +
              f10 * fv * (1.0f - fu)          + f11 * fv * fu;
    }
  }
  beta *= (1.0f / 9.0f);

  const float af = (fmaxf(x2 - x1, 0.0f) * fmaxf(y2 - y1, 0.0f)) / 409600.0f;
  const float bsmall = beta * ((af < 0.01f) ? 1.5f : 1.0f);
  const float calobj = obj[base] - bsmall;

  // class max/argmax over 80 logits: 20 non-temporal b128 loads per anchor
  const vfloat4* __restrict__ row = (const vfloat4*)(cls + base * CN);
  float m = -3.402823466e38f;
  int mi = 0;
#pragma unroll 4
  for (int cb = 0; cb < CN / 4; ++cb) {
    vfloat4 v4 = __builtin_nontemporal_load(row + cb);
    if (v4.x > m) { m = v4.x; mi = 4 * cb + 0; }
    if (v4.y > m) { m = v4.y; mi = 4 * cb + 1; }
    if (v4.z > m) { m = v4.z; mi = 4 * cb + 2; }
    if (v4.w > m) { m = v4.w; mi = 4 * cb + 3; }
  }

  const float pobj = 1.0f / (1.0f + expf(-calobj));
  const float pcls = 1.0f / (1.0f + expf(-(m - 0.5f * bsmall)));
  const float comb = pobj * pcls;
  const float masked = (comb >= 0.25f) ? comb : __uint_as_float(0xFF7FFFFFu); // -FLT_MAX

  const unsigned u = __float_as_uint(masked);
  const unsigned key = (u & 0x80000000u) ? ~u : (u | 0x80000000u); // order-preserving

  okey[base]    = key;
  oclsidx[base] = mi;
  obeta[base]   = bsmall;
}

// ---------------------------------------------------------------------------
// Kernel 2: one block per batch. Stage the 8400 keys into LDS with the
// CDNA5 async-to-LDS path, radix-select the K-th key (4x 8-bit passes over
// LDS; 256-bin histogram reduced to 16 group totals on the matrix pipe with
// V_WMMA_F32_16X16X4_F32 x4, exact for integer counts), compact >T and ==T
// candidates, bitonic-sort 512 (key,~idx) pairs, then gather + write.
// ---------------------------------------------------------------------------
__global__ __launch_bounds__(256) void dpc_topk(
    const unsigned* __restrict__ keys, const int* __restrict__ clsidx,
    const float* __restrict__ beta, const float* __restrict__ boxes,
    float* __restrict__ out)
{
  __shared__ unsigned skeys[AN];                 // 33.6 KB
  __shared__ unsigned hist[256];                 // 1 KB
  __shared__ unsigned sG[16];                    // 16-bin group totals
  __shared__ unsigned sPrefix, sRank;
  __shared__ int cntG, cntE;
  __shared__ unsigned idxG[KN], idxE[KN];        // 2.4 KB
  __shared__ unsigned long long spairs[NSORT];   // 4 KB

  const int tid = threadIdx.x;
  const int b = blockIdx.x;
  const unsigned* __restrict__ gk = keys + (long)b * AN;

  // --- async stage keys -> LDS (2100 b128 transfers), CDNA5 async path ---
  const unsigned ldsbase = (unsigned)(uintptr_t)(&skeys[0]);
  for (int i = tid; i < (AN * 4) / 16; i += 256) {
    const void* g = (const void*)((const char*)gk + i * 16);
    const unsigned dst = ldsbase + (unsigned)(i * 16);
    asm volatile("global_load_async_to_lds_b128 %0, %1, off"
                 :: "v"(dst), "v"(g) : "memory");
  }
  asm volatile("s_wait_asynccnt 0" ::: "memory");
  __syncthreads();

  // --- 4-pass MSB radix select of the K-th largest key ---
  unsigned prefix = 0u, rank = KN;
  for (int pass = 3; pass >= 0; --pass) {
    const int shift = pass * 8;
    hist[tid] = 0u;
    __syncthreads();
    const unsigned pmask = (pass == 3) ? 0u : (0xFFFFFFFFu << ((pass + 1) * 8));
    for (int i = tid; i < AN; i += 256) {
      const unsigned k = skeys[i];
      if ((k & pmask) == prefix)
        atomicAdd(&hist[(k >> shift) & 255u], 1u);
    }
    __syncthreads();

#if USE_WMMA_SCAN
    // Wave 0 (full EXEC, as WMMA requires): view hist as H[16][16],
    // compute row totals G[m] = sum_k H[m][k] via 4 accumulating
    // 16x16x4 f32 WMMAs against an all-ones B (layout-invariant).
    if (tid < 32) {
      const int lane = tid;
      const int half = lane >> 4;       // 0: lanes 0-15, 1: lanes 16-31
      const int msel = lane & 15;       // A-matrix row M = lane % 16
      v8f acc = {};
      const v2f bones = {1.0f, 1.0f};
#pragma unroll
      for (int q = 0; q < 4; ++q) {
        // Documented 16x4 f32 A layout: VGPR g, lane L -> K = g + 2*(L/16)
        v2f afrag;
        afrag.x = (float)hist[16 * msel + 4 * q + 2 * half + 0];
        afrag.y = (float)hist[16 * msel + 4 * q + 2 * half + 1];
        acc = __builtin_amdgcn_wmma_f32_16x16x4_f32(
            /*neg_a=*/false, afrag, /*neg_b=*/false, bones,
            /*c_mod=*/(short)0, acc, /*reuse_a=*/false, /*reuse_b=*/false);
      }
      // D layout: VGPR g holds M = g + 8*half (broadcast over N) -> exact ints
#pragma unroll
      for (int g = 0; g < 8; ++g)
        sG[g + 8 * half] = (unsigned)acc[g];   // 16 lanes write same value
    }
    __syncthreads();
    if (tid == 0) {
      unsigned need = rank;
      int gsel = 0;
      for (int gg = 15; gg >= 0; --gg) {
        const unsigned c = sG[gg];
        if (c >= need) { gsel = gg; break; }
        need -= c;
      }
      int bsel = 16 * gsel;
      for (int bb = 16 * gsel + 15; bb >= 16 * gsel; --bb) {
        const unsigned c = hist[bb];
        if (c >= need) { bsel = bb; break; }
        need -= c;
      }
      sPrefix = prefix | ((unsigned)bsel << shift);
      sRank = need;
    }
#else
    if (tid == 0) {
      unsigned need = rank;
      int bsel = 0;
      for (int bb = 255; bb >= 0; --bb) {
        const unsigned c = hist[bb];
        if (c >= need) { bsel = bb; break; }
        need -= c;
      }
      sPrefix = prefix | ((unsigned)bsel << shift);
      sRank = need;
    }
#endif
    __syncthreads();
    prefix = sPrefix;
    rank = sRank;
    __syncthreads();
  }
  const unsigned T = prefix; // K-th largest key value

  // --- compact candidates: strictly-greater (< K of them) and equal-to-T ---
  if (tid == 0) { cntG = 0; cntE = 0; }
  __syncthreads();
  for (int i = tid; i < AN; i += 256) {
    const unsigned k = skeys[i];
    if (k > T) {
      const int p = atomicAdd(&cntG, 1);
      if (p < KN) idxG[p] = (unsigned)i;
    } else if (k == T) {
      const int p = atomicAdd(&cntE, 1);
      if (p < KN) idxE[p] = (unsigned)i;
    }
  }
  __syncthreads();
  const int nG = min(cntG, KN);

  // --- build 512 sort pairs: (key << 32) | ~idx ; pad with 0 ---
  for (int s = tid; s < NSORT; s += 256) {
    unsigned long long p = 0ull;
    if (s < KN) {
      const unsigned idx = (s < nG) ? idxG[s] : idxE[s - nG];
      const unsigned key = skeys[idx];
      p = ((unsigned long long)key << 32) |
          (unsigned long long)(0xFFFFFFFFu - idx);
    }
    spairs[s] = p;
  }

  // --- bitonic sort, descending ---
  for (int k = 2; k <= NSORT; k <<= 1) {
    for (int j = k >> 1; j > 0; j >>= 1) {
      __syncthreads();
      for (int i = tid; i < NSORT; i += 256) {
        const int ixj = i ^ j;
        if (ixj > i) {
          const unsigned long long va = spairs[i];
          const unsigned long long vb = spairs[ixj];
          const bool descBlock = ((i & k) == 0);
          if (descBlock ? (va < vb) : (va > vb)) {
            spairs[i] = vb;
            spairs[ixj] = va;
          }
        }
      }
    }
  }
  __syncthreads();

  // --- gather + write outputs (concatenated flat: boxes,scores,cls,susp,valid)
  float* __restrict__ obox = out;
  float* __restrict__ oscr = out + (long)BN * KN * 4;
  float* __restrict__ ocls = out + (long)BN * KN * 5;
  float* __restrict__ osus = out + (long)BN * KN * 6;
  float* __restrict__ oval = out + (long)BN * KN * 7;

  for (int s = tid; s < KN; s += 256) {
    const unsigned long long p = spairs[s];
    const unsigned key = (unsigned)(p >> 32);
    const unsigned low = (unsigned)(p & 0xFFFFFFFFull);
    const unsigned idx = 0xFFFFFFFFu - low;
    const unsigned u = (key & 0x80000000u) ? (key & 0x7FFFFFFFu) : ~key;
    const float score = __uint_as_float(u);
    const bool valid = (score >= 0.25f); // false for -FLT_MAX and NaN padding

    vfloat4 bb = {0.0f, 0.0f, 0.0f, 0.0f};
    float cl = 0.0f, su = 0.0f, sc = 0.0f;
    if (valid) {
      const long g = (long)b * AN + (long)idx;
      bb = *(const vfloat4*)(boxes + 4 * g);
      cl = (float)clsidx[g];
      su = beta[g];
      sc = score;
    }
    const long o = (long)b * KN + s;
    *(vfloat4*)(obox + 4 * o) = bb;
    oscr[o] = sc;
    ocls[o] = cl;
    osus[o] = su;
    oval[o] = valid ? 1.0f : 0.0f;
  }
}

// ---------------------------------------------------------------------------
extern "C" void kernel_launch(void* const* d_in, const int* in_sizes, int n_in,
                              void* d_out, int out_size, void* d_ws, size_t ws_size,
                              hipStream_t stream)
{
  (void)in_sizes; (void)n_in; (void)out_size; (void)ws_size;

  const float* boxes = (const float*)d_in[0];
  const float* obj   = (const float*)d_in[1];
  const float* cls   = (const float*)d_in[2];
  const float* field = (const float*)d_in[3];

  // workspace: keys[B*A] u32 | bestcls[B*A] i32 | beta_small[B*A] f32  (~6.5 MB)
  unsigned* wkeys = (unsigned*)d_ws;
  int*      wcls  = (int*)(wkeys + (size_t)BN * AN);
  float*    wbeta = (float*)(wcls + (size_t)BN * AN);

  dim3 g1((AN + 255) / 256, BN);
  dpc_score<<<g1, 256, 0, stream>>>(boxes, obj, cls, field, wkeys, wcls, wbeta);
  dpc_topk<<<BN, 256, 0, stream>>>(wkeys, wcls, wbeta, boxes, (float*)d_out);
}